// GraphConv_56556129354466
// MI455X (gfx1250) — compile-verified
//
#include <hip/hip_runtime.h>

#define NODES 100000
#define EDGES 1600000
#define DIM 128
#define A_STRIDE 132   // 128 + 4 pad: row n lands on bank (n*132)%64 -> conflict-free

typedef __attribute__((ext_vector_type(2))) float v2f;
typedef __attribute__((ext_vector_type(8))) float v8f;

// ---------------------------------------------------------------- zero buffers
__global__ void gc_zero(float* __restrict__ out, float* __restrict__ degs) {
    int i = blockIdx.x * blockDim.x + threadIdx.x;
    if (i < NODES * DIM) out[i] = 0.0f;
    if (i < 2 * NODES)   degs[i] = 0.0f;
}

// ---------------------------------------------------------------- degree count
__global__ void gc_degree(const int* __restrict__ src, const int* __restrict__ dst,
                          float* __restrict__ outdeg, float* __restrict__ indeg) {
    int e = blockIdx.x * blockDim.x + threadIdx.x;
    if (e >= EDGES) return;
    atomicAdd(&outdeg[src[e]], 1.0f);
    atomicAdd(&indeg[dst[e]], 1.0f);
}

// ---------------------------------------------------------------- deg -> rsqrt(max(deg,1))
__global__ void gc_norm(float* __restrict__ degs) {
    int i = blockIdx.x * blockDim.x + threadIdx.x;
    if (i >= 2 * NODES) return;
    float d = degs[i];
    degs[i] = rsqrtf(d < 1.0f ? 1.0f : d);
}

// ---------------------------------------------------------------- edge scatter
// one wave32 per edge: lane l moves float4 (cols 4l..4l+3) of feat[src],
// scaled by out_norm[src], atomically added into agg[dst].
__global__ void gc_scatter(const float* __restrict__ feat, const int* __restrict__ src,
                           const int* __restrict__ dst, const float* __restrict__ out_norm,
                           float* __restrict__ agg) {
    int gid  = blockIdx.x * blockDim.x + threadIdx.x;
    int e    = gid >> 5;
    int lane = gid & 31;
    if (e >= EDGES) return;
    int s = src[e];
    int d = dst[e];
    float nrm = out_norm[s];
    float4 v = ((const float4*)(feat + (size_t)s * DIM))[lane];
    float* a = agg + (size_t)d * DIM + lane * 4;
    atomicAdd(a + 0, v.x * nrm);
    atomicAdd(a + 1, v.y * nrm);
    atomicAdd(a + 2, v.z * nrm);
    atomicAdd(a + 3, v.w * nrm);
}

// ---------------------------------------------------------------- in-place GEMM
// out[m0+0..15, :] = (agg[m0+0..15, :] * in_norm) @ W + b  via v_wmma_f32_16x16x4_f32.
// 256 threads = 8 waves; wave w owns output columns [16w, 16w+16).
__global__ void __launch_bounds__(256, 1)
gc_gemm(float* __restrict__ inout, const float* __restrict__ W,
        const float* __restrict__ bias, const float* __restrict__ in_norm) {
    __shared__ float At[16 * A_STRIDE];

    const int tid = threadIdx.x;
    const int m0  = blockIdx.x * 16;

    // Stage the 16x128 A-tile (scaled by in_norm) into LDS, coalesced.
    for (int i = tid; i < 16 * DIM; i += 256) {
        int r = i >> 7;
        int c = i & 127;
        At[r * A_STRIDE + c] = inout[(size_t)(m0 + r) * DIM + c] * in_norm[m0 + r];
    }

    const int wv   = tid >> 5;
    const int lane = tid & 31;
    const int n0   = wv * 16;
    const int nl   = lane & 15;
    const int hi   = lane >> 4;          // lanes 16-31 hold K+2/K+3 halves

    // Preload this wave's B fragments (W cols n0..n0+15, all K) into registers.
    // ISA B layout for 16x16x4: VGPR0 = {K, K+2}, VGPR1 = {K+1, K+3} across lane halves.
    v2f breg[32];
#pragma unroll
    for (int s = 0; s < 32; ++s) {
        int k = 4 * s + 2 * hi;
        v2f t;
        t.x = W[(size_t)k       * DIM + n0 + nl];
        t.y = W[(size_t)(k + 1) * DIM + n0 + nl];
        breg[s] = t;
    }

    // C initialized to broadcast bias (every C VGPR = b[n] for this lane's column).
    v8f acc;
    float bb = bias[n0 + nl];
#pragma unroll
    for (int i = 0; i < 8; ++i) acc[i] = bb;

    __syncthreads();

    // K-loop: 32 chained WMMAs, A fragments streamed from LDS (ds_load_b64).
    const float* arow = At + nl * A_STRIDE;
#pragma unroll
    for (int s = 0; s < 32; ++s) {
        v2f a = *(const v2f*)(arow + 4 * s + 2 * hi);
        acc = __builtin_amdgcn_wmma_f32_16x16x4_f32(
            /*neg_a=*/false, a, /*neg_b=*/false, breg[s],
            /*c_mod=*/(short)0, acc, /*reuse_a=*/false, /*reuse_b=*/false);
    }

    // D layout: VGPR i = row m0+i (lanes 0-15) / row m0+8+i (lanes 16-31), col n0+nl.
#pragma unroll
    for (int i = 0; i < 8; ++i) {
        inout[(size_t)(m0 + i + 8 * hi) * DIM + n0 + nl] = acc[i];
    }
}

// ---------------------------------------------------------------- launch
extern "C" void kernel_launch(void* const* d_in, const int* in_sizes, int n_in,
                              void* d_out, int out_size, void* d_ws, size_t ws_size,
                              hipStream_t stream) {
    const float* feat = (const float*)d_in[0];
    const int*   src  = (const int*)d_in[1];
    const int*   dst  = (const int*)d_in[2];
    const float* W    = (const float*)d_in[3];
    const float* bias = (const float*)d_in[4];
    float* out  = (float*)d_out;

    float* degs = (float*)d_ws;          // [0..N) out-deg/norm, [N..2N) in-deg/norm
    float* outn = degs;
    float* inn  = degs + NODES;

    int totz = NODES * DIM;
    gc_zero   <<<(totz + 255) / 256, 256, 0, stream>>>(out, degs);
    gc_degree <<<(EDGES + 255) / 256, 256, 0, stream>>>(src, dst, outn, inn);
    gc_norm   <<<(2 * NODES + 255) / 256, 256, 0, stream>>>(degs);

    long sthreads = (long)EDGES * 32;
    gc_scatter<<<(unsigned)((sthreads + 255) / 256), 256, 0, stream>>>(feat, src, dst, outn, out);

    gc_gemm   <<<NODES / 16, 256, 0, stream>>>(out, W, bias, inn);
}